// MambaLayer_27367531610163
// MI455X (gfx1250) — compile-verified
//
#include <hip/hip_runtime.h>
#include <hip/hip_bf16.h>

// ---- problem constants (from reference) ----
#define BB 4
#define LL 1024
#define DM 512
#define DS 16
#define DC 4
#define DI 1024
#define MROWS (BB * LL)   // 4096

typedef __attribute__((ext_vector_type(2))) float v2f;
typedef __attribute__((ext_vector_type(8))) float v8f;

// Async global->LDS (CDNA5): use if this toolchain declares the builtins.
#if defined(__has_builtin)
#if __has_builtin(__builtin_amdgcn_global_load_async_to_lds_b128) && \
    __has_builtin(__builtin_amdgcn_s_wait_asynccnt)
#define USE_ASYNC_LDS 1
#endif
#endif

#ifdef USE_ASYNC_LDS
// builtin prototype (from clang diagnostic): params are pointers to a
// 16-byte int vector in global (as1) / LDS (as3) address spaces.
typedef int async_v4i __attribute__((vector_size(16)));
typedef __attribute__((address_space(1))) async_v4i* async_gptr;
typedef __attribute__((address_space(3))) async_v4i* async_lptr;
typedef __attribute__((address_space(1))) void* as1_void;
typedef __attribute__((address_space(3))) void* as3_void;
#define ASYNC_CP_B128(gp, lp)                                            \
    __builtin_amdgcn_global_load_async_to_lds_b128(                      \
        (async_gptr)(as1_void)(gp), (async_lptr)(as3_void)(lp), 0, 0)
#endif

// =====================================================================
// Generic fp32 WMMA GEMM: C[M,N] = A[M,K] @ B[K,N] (+ bias[N] optional)
// Block tile 128x64, 8 waves (wave32), wave tile 32x32 = 2x2 frags of 16x16.
// V_WMMA_F32_16X16X4_F32 (fp32 A/B, f32 accum) to match fp32 reference.
// Double-buffered LDS; async global->LDS copies when available.
// =====================================================================
#define GBM 128
#define GBN 64
#define GBK 16
#define GBKP 20   // padded A stride (floats): 16B-aligned b128 chunks, distinct banks
#define GBNP 80   // padded B stride (floats): lanes 0-15 / 16-31 hit disjoint banks

__global__ __launch_bounds__(256) void gemm_f32_wmma_kernel(
    const float* __restrict__ A, const float* __restrict__ Bm,
    float* __restrict__ C, const float* __restrict__ bias,
    int Mdim, int Ndim, int Kdim)
{
    __shared__ __align__(16) float sA[2][GBM * GBKP];
    __shared__ __align__(16) float sB[2][GBK * GBNP];

    const int tid  = threadIdx.x;
    const int lane = tid & 31;
    const int wave = tid >> 5;          // 0..7
    const int wm   = wave & 3;          // wave row (M dir) 0..3
    const int wn   = wave >> 2;         // wave col (N dir) 0..1
    const int l16  = lane & 15;
    const int half = lane >> 4;         // 0: K pair 0/1, 1: K pair 2/3

    const int blockM = blockIdx.y * GBM;
    const int blockN = blockIdx.x * GBN;

    // stage one 128x16 A tile + 16x64 B tile into LDS buffer `buf`
    auto stage = [&](int buf, int k0) {
#pragma unroll
        for (int r = 0; r < 2; ++r) {
            int idx = tid + r * 256;            // float4 index over A tile
            int row = idx >> 2;                 // 4 float4 per row
            int c4  = (idx & 3) << 2;
            const float* gp = A + (size_t)(blockM + row) * Kdim + k0 + c4;
            float*       lp = &sA[buf][row * GBKP + c4];
#ifdef USE_ASYNC_LDS
            ASYNC_CP_B128(gp, lp);
#else
            const float4 v = *(const float4*)gp;
            lp[0] = v.x; lp[1] = v.y; lp[2] = v.z; lp[3] = v.w;
#endif
        }
        {
            int row  = tid >> 4;                // 0..15
            int c4   = (tid & 15) << 2;         // 0..60
            int gcol = blockN + c4;
            const float* gp = Bm + (size_t)(k0 + row) * Ndim + gcol;
            float*       lp = &sB[buf][row * GBNP + c4];
#ifdef USE_ASYNC_LDS
            // out-of-range columns only feed output cols >= N (discarded on
            // store), so stale LDS there is harmless -- skip the load only.
            if (gcol < Ndim)
                ASYNC_CP_B128(gp, lp);
#else
            float4 v = make_float4(0.f, 0.f, 0.f, 0.f);
            if (gcol < Ndim) v = *(const float4*)gp;
            lp[0] = v.x; lp[1] = v.y; lp[2] = v.z; lp[3] = v.w;
#endif
        }
    };

    v8f acc[2][2];
#pragma unroll
    for (int mi = 0; mi < 2; ++mi)
#pragma unroll
        for (int ni = 0; ni < 2; ++ni)
            acc[mi][ni] = (v8f){0.f, 0.f, 0.f, 0.f, 0.f, 0.f, 0.f, 0.f};

    // prologue: fill buffer 0
    stage(0, 0);
#ifdef USE_ASYNC_LDS
    __builtin_amdgcn_s_wait_asynccnt(0);
#endif
    __syncthreads();

    int buf = 0;
    for (int k0 = 0; k0 < Kdim; k0 += GBK) {
        // kick off next tile's copies before computing (overlap with WMMA)
        if (k0 + GBK < Kdim) stage(buf ^ 1, k0 + GBK);
        // light prefetch of the tile after next into caches
        if (k0 + 2 * GBK < Kdim)
            __builtin_prefetch(A + (size_t)(blockM + (tid >> 1)) * Kdim + k0 + 2 * GBK, 0, 1);

#pragma unroll
        for (int kk = 0; kk < GBK; kk += 4) {
            // A frags: lane<16 -> (K=kk,kk+1) of row M=l16; lane>=16 -> (kk+2,kk+3)
            v2f af[2];
#pragma unroll
            for (int mi = 0; mi < 2; ++mi) {
                int m = wm * 32 + mi * 16 + l16;
                af[mi] = *(const v2f*)(&sA[buf][m * GBKP + kk + half * 2]);
            }
            // B frags: lane<16 -> rows K=kk,kk+1 of col N=l16; lane>=16 -> kk+2,kk+3
            v2f bf[2];
#pragma unroll
            for (int ni = 0; ni < 2; ++ni) {
                int n    = wn * 32 + ni * 16 + l16;
                int krow = kk + half * 2;
                v2f b;
                b.x = sB[buf][(krow + 0) * GBNP + n];
                b.y = sB[buf][(krow + 1) * GBNP + n];
                bf[ni] = b;
            }
#pragma unroll
            for (int mi = 0; mi < 2; ++mi)
#pragma unroll
                for (int ni = 0; ni < 2; ++ni)
                    acc[mi][ni] = __builtin_amdgcn_wmma_f32_16x16x4_f32(
                        false, af[mi], false, bf[ni], (short)0, acc[mi][ni], false, false);
        }

#ifdef USE_ASYNC_LDS
        __builtin_amdgcn_s_wait_asynccnt(0);
#endif
        __syncthreads();   // next buffer ready AND current buffer free
        buf ^= 1;
    }

    // ---- store: D[m][n], m = r + 8*half, n = l16 within each 16x16 tile ----
#pragma unroll
    for (int mi = 0; mi < 2; ++mi) {
#pragma unroll
        for (int ni = 0; ni < 2; ++ni) {
            int rowBase = blockM + wm * 32 + mi * 16 + half * 8;
            int col     = blockN + wn * 32 + ni * 16 + l16;
            if (col < Ndim) {
                float bv = bias ? bias[col] : 0.f;
#pragma unroll
                for (int r = 0; r < 8; ++r)
                    C[(size_t)(rowBase + r) * Ndim + col] = acc[mi][ni][r] + bv;
            }
        }
    }
}

// =====================================================================
// clip x to [-10,10]
// =====================================================================
__global__ void clip_kernel(const float* __restrict__ x, float* __restrict__ xc, int n)
{
    int i = blockIdx.x * blockDim.x + threadIdx.x;
    if (i < n) xc[i] = fminf(fmaxf(x[i], -10.f), 10.f);
}

// =====================================================================
// depthwise causal conv1d (k=4, left pad 3) + bias + SiLU
// input xp = first DI cols of y1 (row stride 2*DI)
// =====================================================================
__global__ void conv_silu_kernel(const float* __restrict__ y1,
                                 const float* __restrict__ convw,
                                 const float* __restrict__ convb,
                                 float* __restrict__ xc)
{
    int idx = blockIdx.x * blockDim.x + threadIdx.x;   // over B*L*DI
    if (idx >= BB * LL * DI) return;
    int i = idx % DI;
    int t = (idx / DI) % LL;
    int b = idx / (DI * LL);
    const float* base = y1 + (size_t)b * LL * (2 * DI);
    float acc = convb[i];
#pragma unroll
    for (int h = 0; h < DC; ++h) {
        int tt = t - (DC - 1) + h;
        if (tt >= 0) acc = fmaf(convw[i * DC + h], base[(size_t)tt * (2 * DI) + i], acc);
    }
    xc[idx] = acc / (1.f + expf(-acc));   // silu
}

// =====================================================================
// selective scan: one thread per (b, inner channel), 16 states in regs.
// B_t/C_t (32 floats per step) staged in LDS per 32-step chunk.
// =====================================================================
__global__ __launch_bounds__(256) void scan_kernel(
    const float* __restrict__ xc, const float* __restrict__ delta,
    const float* __restrict__ ssm, const float* __restrict__ A_log,
    float* __restrict__ ys)
{
    __shared__ float sBC[32 * 32];   // [t_local][32]: 16 B, then 16 C
    const int tid = threadIdx.x;
    const int b   = blockIdx.x >> 2;            // 4 blocks per batch
    const int i   = ((blockIdx.x & 3) << 8) + tid;

    float Al[DS];
#pragma unroll
    for (int s = 0; s < DS; ++s) Al[s] = A_log[s];
    float h[DS];
#pragma unroll
    for (int s = 0; s < DS; ++s) h[s] = 0.f;

    const float* xcb = xc    + (size_t)b * LL * DI + i;
    const float* dtb = delta + (size_t)b * LL * DI + i;
    const float* ssb = ssm   + (size_t)b * LL * 32;
    float*       ysb = ys    + (size_t)b * LL * DI + i;

    for (int t0 = 0; t0 < LL; t0 += 32) {
        __syncthreads();
#pragma unroll
        for (int r = 0; r < 4; ++r) {
            int fi = tid + r * 256;              // 1024 floats per chunk
            sBC[fi] = ssb[(size_t)t0 * 32 + fi];
        }
        __syncthreads();
        for (int tl = 0; tl < 32; ++tl) {
            const int t = t0 + tl;
            const float u = xcb[(size_t)t * DI];
            const float d = dtb[(size_t)t * DI];
            const float* Bs = &sBC[tl * 32];
            const float* Cs = Bs + 16;
            float y = 0.f;
#pragma unroll
            for (int s = 0; s < DS; ++s) {
                float z = fminf(fmaxf(d * Al[s], -5.f), 5.f);
                float dA = expf(z);
                h[s] = fmaf(dA, h[s], (d * Bs[s]) * u);
                y = fmaf(h[s], Cs[s], y);
            }
            ysb[(size_t)t * DI] = y;
        }
    }
}

// =====================================================================
// y = (ys + xc*D_skip) * silu(res);  res = second half of y1 row
// =====================================================================
__global__ void gate_kernel(float* __restrict__ ys, const float* __restrict__ xc,
                            const float* __restrict__ y1, const float* __restrict__ Dskip)
{
    int idx = blockIdx.x * blockDim.x + threadIdx.x;   // over B*L*DI
    if (idx >= BB * LL * DI) return;
    int i   = idx % DI;
    int row = idx / DI;
    float r   = y1[(size_t)row * (2 * DI) + DI + i];
    float sil = r / (1.f + expf(-r));
    ys[idx] = (ys[idx] + xc[idx] * Dskip[i]) * sil;
}

// =====================================================================
// LayerNorm over D_MODEL=512 per row: r = outmm + xclip
// =====================================================================
__global__ __launch_bounds__(256) void ln_kernel(
    const float* __restrict__ outmm, const float* __restrict__ xclip,
    const float* __restrict__ gamma, const float* __restrict__ beta,
    float* __restrict__ out)
{
    __shared__ float rsum[8], rsq[8];
    __shared__ float sMu, sInv;
    const int row  = blockIdx.x;
    const int tid  = threadIdx.x;
    const size_t base = (size_t)row * DM;

    float r0 = outmm[base + tid]       + xclip[base + tid];
    float r1 = outmm[base + 256 + tid] + xclip[base + 256 + tid];
    float sum = r0 + r1;
    float sq  = r0 * r0 + r1 * r1;
#pragma unroll
    for (int off = 16; off > 0; off >>= 1) {
        sum += __shfl_xor(sum, off, 32);
        sq  += __shfl_xor(sq,  off, 32);
    }
    if ((tid & 31) == 0) { rsum[tid >> 5] = sum; rsq[tid >> 5] = sq; }
    __syncthreads();
    if (tid == 0) {
        float S = 0.f, Q = 0.f;
        for (int w = 0; w < 8; ++w) { S += rsum[w]; Q += rsq[w]; }
        float mu  = S * (1.f / DM);
        float var = Q * (1.f / DM) - mu * mu;
        sMu  = mu;
        sInv = rsqrtf(var + 1e-5f);
    }
    __syncthreads();
    out[base + tid]       = gamma[tid]       * (r0 - sMu) * sInv + beta[tid];
    out[base + 256 + tid] = gamma[tid + 256] * (r1 - sMu) * sInv + beta[tid + 256];
}

// =====================================================================
// launch
// =====================================================================
extern "C" void kernel_launch(void* const* d_in, const int* in_sizes, int n_in,
                              void* d_out, int out_size, void* d_ws, size_t ws_size,
                              hipStream_t stream)
{
    const float* x      = (const float*)d_in[0];
    const float* W_in   = (const float*)d_in[1];
    const float* conv_w = (const float*)d_in[2];
    const float* conv_b = (const float*)d_in[3];
    const float* W_x    = (const float*)d_in[4];
    const float* W_dt   = (const float*)d_in[5];
    const float* b_dt   = (const float*)d_in[6];
    const float* A_log  = (const float*)d_in[7];
    const float* D_skip = (const float*)d_in[8];
    const float* W_out  = (const float*)d_in[9];
    const float* gamma  = (const float*)d_in[10];
    const float* beta   = (const float*)d_in[11];

    float* ws = (float*)d_ws;
    const size_t OFF_XCLIP = 0;                                   // 4096*512
    const size_t OFF_Y1    = OFF_XCLIP + (size_t)MROWS * DM;      // 4096*2048
    const size_t OFF_XC    = OFF_Y1    + (size_t)MROWS * 2 * DI;  // 4096*1024
    const size_t OFF_SSM   = OFF_XC    + (size_t)MROWS * DI;      // 4096*32
    const size_t OFF_DELTA = OFF_SSM   + (size_t)MROWS * 32;      // 4096*1024
    const size_t OFF_YS    = OFF_DELTA + (size_t)MROWS * DI;      // 4096*1024
    const size_t OFF_OUT   = OFF_DELTA;                           // delta dead after scan

    float* xclip = ws + OFF_XCLIP;
    float* y1    = ws + OFF_Y1;
    float* xc    = ws + OFF_XC;
    float* ssm   = ws + OFF_SSM;
    float* delta = ws + OFF_DELTA;
    float* ys    = ws + OFF_YS;
    float* outmm = ws + OFF_OUT;

    // 1) clip
    clip_kernel<<<(MROWS * DM + 255) / 256, 256, 0, stream>>>(x, xclip, MROWS * DM);
    // 2) x_and_res = xclip @ W_in   [4096 x 2048 x 512]
    gemm_f32_wmma_kernel<<<dim3(2 * DI / GBN, MROWS / GBM), 256, 0, stream>>>(
        xclip, W_in, y1, nullptr, MROWS, 2 * DI, DM);
    // 3) depthwise causal conv + silu -> xc
    conv_silu_kernel<<<(MROWS * DI + 255) / 256, 256, 0, stream>>>(y1, conv_w, conv_b, xc);
    // 4) ssm = xc @ W_x   [4096 x 32 x 1024]
    gemm_f32_wmma_kernel<<<dim3(1, MROWS / GBM), 256, 0, stream>>>(
        xc, W_x, ssm, nullptr, MROWS, 2 * DS, DI);
    // 5) delta = xc @ W_dt + b_dt   [4096 x 1024 x 1024]
    gemm_f32_wmma_kernel<<<dim3(DI / GBN, MROWS / GBM), 256, 0, stream>>>(
        xc, W_dt, delta, b_dt, MROWS, DI, DI);
    // 6) selective scan
    scan_kernel<<<BB * (DI / 256), 256, 0, stream>>>(xc, delta, ssm, A_log, ys);
    // 7) D-skip + gating (in place on ys)
    gate_kernel<<<(MROWS * DI + 255) / 256, 256, 0, stream>>>(ys, xc, y1, D_skip);
    // 8) out = ys @ W_out   [4096 x 512 x 1024]
    gemm_f32_wmma_kernel<<<dim3(DM / GBN, MROWS / GBM), 256, 0, stream>>>(
        ys, W_out, outmm, nullptr, MROWS, DM, DI);
    // 9) residual + LayerNorm -> d_out
    ln_kernel<<<MROWS, 256, 0, stream>>>(outmm, xclip, gamma, beta, (float*)d_out);
}